// QueryAndGroup_20486994002139
// MI455X (gfx1250) — compile-verified
//
#include <hip/hip_runtime.h>

typedef float v2f __attribute__((ext_vector_type(2)));
typedef float v8f __attribute__((ext_vector_type(8)));
typedef unsigned int v4u __attribute__((ext_vector_type(4)));
typedef int v8i __attribute__((ext_vector_type(8)));
typedef int v4i __attribute__((ext_vector_type(4)));

#define BQ      4
#define NSUP    16384
#define NQ      4096
#define NCH     64
#define NS      32
#define R2      0.01f
#define CHUNK   1024
#define NWAVE   4
#define QPB     64                  // queries per block = NWAVE * 16
#define NCHUNKS (NSUP / CHUNK)      // 16
#define CH_FLOATS (CHUNK * 3)       // 3072 dwords = 12 KB per chunk

#if defined(__has_builtin)
#if __has_builtin(__builtin_amdgcn_tensor_load_to_lds) && \
    __has_builtin(__builtin_amdgcn_s_wait_tensorcnt)
#define USE_TDM 1
#endif
#endif

#if defined(USE_TDM)
// 1-D TDM copy: ndw dwords from global to LDS.  D# bitfields per CDNA5 ISA
// 8.3-8.5: group0 = {count/lds_addr/global_addr/type}, group1 = dims/strides.
// 6-arg builtin form: (v4u g0, v8i g1, v4i g2, v4i g3, v8i pad, i32 cpol)
__device__ __forceinline__ void tdm_load_1d(const float* gsrc, void* ldst, int ndw) {
    unsigned long long ga = (unsigned long long)(uintptr_t)gsrc;
    unsigned lds = (unsigned)(uintptr_t)ldst;   // LDS byte address = addr[31:0]
    v4u g0;
    g0[0] = 1u;                                       // count=1, user mode
    g0[1] = lds;                                      // lds_addr [63:32]
    g0[2] = (unsigned)(ga & 0xFFFFFFFFu);             // global_addr[31:0]
    g0[3] = (unsigned)((ga >> 32) & 0x1FFFFFFu)       // global_addr[56:32]
          | (2u << 30);                               // type=2 ("image")
    v8i g1;
    g1[0] = (int)(2u << 16);                          // data_size=2 (4B), wg_mask=0
    g1[1] = (int)(((unsigned)ndw & 0xFFFFu) << 16);   // tensor_dim0[15:0]
    g1[2] = (int)((((unsigned)ndw >> 16) & 0xFFFFu)   // tensor_dim0[31:16]
          | (1u << 16));                              // tensor_dim1 = 1
    g1[3] = (int)(((unsigned)ndw & 0xFFFFu) << 16);   // tile_dim0 = ndw
    g1[4] = 1;                                        // tile_dim1=1, tile_dim2=0
    g1[5] = ndw;                                      // tensor_dim0_stride[31:0]
    g1[6] = 0;
    g1[7] = 0;
    v4i gz4 = {0, 0, 0, 0};                           // groups 2/3 unused (<=2D)
    v8i gz8 = {0, 0, 0, 0, 0, 0, 0, 0};
    __builtin_amdgcn_tensor_load_to_lds(g0, g1, gz4, gz4, gz8, 0);
}
#endif

__global__ __launch_bounds__(128) void ballq_group_wmma_kernel(
    const float* __restrict__ query_xyz,    // (B, NQ, 3)
    const float* __restrict__ support_xyz,  // (B, NSUP, 3)
    const float* __restrict__ features,     // (B, NCH, NSUP)
    float* __restrict__ out)                // xyz (B,3,NQ,NS) ++ feat (B,NCH,NQ,NS)
{
    __shared__ float raw[2][CH_FLOATS];     // double-buffered raw xyz chunks (24 KB)
    __shared__ int idxLDS[NWAVE][16][NS];
    __shared__ int doneCount;

    const int tid  = threadIdx.x;
    const int lane = tid & 31;
    const int w    = tid >> 5;
    const int blocksPerBatch = NQ / QPB;    // 64
    const int b      = blockIdx.x / blocksPerBatch;
    const int qblock = (blockIdx.x % blocksPerBatch) * QPB;

    const float* supp_b = support_xyz + (size_t)b * NSUP * 3;

    // ---------------- WMMA A and C operand setup (per wave: 16 queries) -----
    // acc[M][n] = q2[M] - 2 q_M . x_n   (C = q2, B row K3 = 0)
    // in-ball test: acc < R2 - |x_n|^2  (per-lane threshold, n = lane & 15)
    const int m = lane & 15;
    const bool lo = (lane < 16);
    const int qrow = qblock + w * 16 + m;
    const float qx = query_xyz[((size_t)b * NQ + qrow) * 3 + 0];
    const float qy = query_xyz[((size_t)b * NQ + qrow) * 3 + 1];
    const float qz = query_xyz[((size_t)b * NQ + qrow) * 3 + 2];
    const float q2 = qx * qx + qy * qy + qz * qz;

    v2f a;
    a[0] = lo ? (-2.0f * qx) : (-2.0f * qz);   // K0 | K2
    a[1] = lo ? (-2.0f * qy) : 0.0f;           // K1 | K3 (x B[3]=0)

    v8f cinit;
    const int hi8 = (lane >> 4) << 3;
    #pragma unroll
    for (int r = 0; r < 8; ++r)
        cinit[r] = __shfl(q2, r + hi8, 32);

    int  cnt      = lo ? 0 : NS;            // lanes >=16 report "done"
    int  firstIdx = 0;                      // empty ball pads with 0
    bool wave_done = false;

#if defined(USE_TDM)
    if (w == 0) tdm_load_1d(supp_b, &raw[0][0], CH_FLOATS);
#endif

    // ---------------- scan support points in index order --------------------
    for (int chunk = 0; chunk < NCHUNKS; ++chunk) {
        const int buf = chunk & 1;
#if defined(USE_TDM)
        if (w == 0) __builtin_amdgcn_s_wait_tensorcnt(0);
        __syncthreads();                    // raw[buf] ready; raw[buf^1] free
        if (w == 0 && chunk + 1 < NCHUNKS)  // async prefetch next chunk via TDM
            tdm_load_1d(supp_b + (size_t)(chunk + 1) * CH_FLOATS,
                        &raw[buf ^ 1][0], CH_FLOATS);
#else
        {
            const float4* gsrc = (const float4*)(supp_b + (size_t)chunk * CH_FLOATS);
            float4* dst = (float4*)&raw[buf][0];
            for (int i = tid; i < CH_FLOATS / 4; i += 128) dst[i] = gsrc[i];
        }
        __syncthreads();
#endif

        if (!wave_done) {  // wave-uniform: EXEC all-ones for WMMA
            const float* rb = &raw[buf][0];
            // software pipeline: LDS loads for tile t+1 issued before ballots of t
            float nx = rb[3 * m], ny = rb[3 * m + 1], nz = rb[3 * m + 2];
            for (int t = 0; t < CHUNK / 16; ++t) {
                const float x = nx, y = ny, z = nz;
                const int pn = (((t + 1) & (CHUNK / 16 - 1)) << 4) + m;
                nx = rb[3 * pn]; ny = rb[3 * pn + 1]; nz = rb[3 * pn + 2];

                const float x2 = x * x + y * y + z * z;
                const float thresh = R2 - x2;
                v2f bb;
                bb[0] = lo ? x : z;         // K0 | K2
                bb[1] = lo ? y : 0.0f;      // K1 | K3 = 0

                v8f acc = __builtin_amdgcn_wmma_f32_16x16x4_f32(
                    false, a, false, bb, (short)0, cinit, false, false);

                unsigned bal[8];
                #pragma unroll
                for (int r = 0; r < 8; ++r)
                    bal[r] = __builtin_amdgcn_ballot_w32(acc[r] < thresh);

                if (lo) {  // lane m owns query row m
                    const int rr = m & 7;
                    unsigned sel =
                        (rr < 4) ? ((rr < 2) ? (rr == 0 ? bal[0] : bal[1])
                                             : (rr == 2 ? bal[2] : bal[3]))
                                 : ((rr < 6) ? (rr == 4 ? bal[4] : bal[5])
                                             : (rr == 6 ? bal[6] : bal[7]));
                    unsigned mask16 = (m < 8) ? (sel & 0xFFFFu) : (sel >> 16);
                    const int base = chunk * CHUNK + t * 16;
                    while (mask16 && cnt < NS) {
                        const int bit = __builtin_ctz(mask16);
                        mask16 &= mask16 - 1u;
                        const int sidx = base + bit;
                        if (cnt == 0) firstIdx = sidx;
                        idxLDS[w][m][cnt] = sidx;
                        ++cnt;
                    }
                }
                if (__builtin_amdgcn_ballot_w32(cnt >= NS) == 0xFFFFFFFFu) {
                    wave_done = true;
                    break;
                }
            }
        }

        // block-wide early-exit vote
        if (tid == 0) doneCount = 0;
        __syncthreads();
        if (lane == 0 && wave_done) atomicAdd(&doneCount, 1);
        __syncthreads();
        if (doneCount == NWAVE) break;
    }

    // pad remaining slots with first-found index (0 if ball empty)
    if (lo) {
        for (int s = cnt; s < NS; ++s) idxLDS[w][m][s] = firstIdx;
    }
    __syncthreads();

    // ---------------- fused gather: grouped_xyz and grouped_features --------
    float* out_xyz  = out;
    float* out_feat = out + (size_t)BQ * 3 * NQ * NS;
    const float* feat_b = features + (size_t)b * NCH * NSUP;
    const int s = lane;  // sample dim -> coalesced stores

    for (int ql = w; ql < QPB; ql += NWAVE) {
        const int qg  = qblock + ql;
        const int idx = idxLDS[ql >> 4][ql & 15][s];

        const float* pp = supp_b + (size_t)idx * 3;
        const float* qp = query_xyz + ((size_t)b * NQ + qg) * 3;
        #pragma unroll
        for (int d = 0; d < 3; ++d) {
            out_xyz[(((size_t)b * 3 + d) * NQ + qg) * NS + s] = pp[d] - qp[d];
        }
        #pragma unroll 4
        for (int c = 0; c < NCH; ++c) {
            out_feat[(((size_t)b * NCH + c) * NQ + qg) * NS + s] =
                feat_b[(size_t)c * NSUP + idx];
        }
    }
}

extern "C" void kernel_launch(void* const* d_in, const int* in_sizes, int n_in,
                              void* d_out, int out_size, void* d_ws, size_t ws_size,
                              hipStream_t stream) {
    const float* query    = (const float*)d_in[0];  // (B, NQ, 3)
    const float* support  = (const float*)d_in[1];  // (B, NSUP, 3)
    const float* features = (const float*)d_in[2];  // (B, NCH, NSUP)
    float* out = (float*)d_out;

    dim3 grid(BQ * (NQ / QPB));   // 256 blocks x 128 threads
    ballq_group_wmma_kernel<<<grid, 128, 0, stream>>>(query, support, features, out);
}